// MixturePredictor_90701119357624
// MI455X (gfx1250) — compile-verified
//
#include <hip/hip_runtime.h>
#include <hip/hip_bf16.h>
#include <math.h>

#define N_NODES    1000000
#define N_EDGES    16000000
#define NUM_GRAPHS 32768
#define IN_DIM     64
#define EMB        32
#define NUM_CLASSES 109

typedef __attribute__((ext_vector_type(2))) float v2f;
typedef __attribute__((ext_vector_type(8))) float v8f;

// Native f32 atomic add (global_atomic_add_f32), avoids CAS loop.
__device__ __forceinline__ void atomAddF32(float* p, float v) {
  unsafeAtomicAdd(p, v);
}

// ---------------------------------------------------------------- utilities
__global__ void k_zero(float* __restrict__ p, int n) {
  int i = blockIdx.x * blockDim.x + threadIdx.x;
  if (i < n) p[i] = 0.0f;
}

__global__ void k_init_deg(float* __restrict__ deg) {
  int i = blockIdx.x * blockDim.x + threadIdx.x;
  if (i < N_NODES) deg[i] = 1.0f;  // self-loop contributes 1
}

__global__ void k_deg_accum(const long long* __restrict__ ei, float* __restrict__ deg) {
  int e = blockIdx.x * blockDim.x + threadIdx.x;
  if (e < N_EDGES) {
    int d = (int)ei[(long long)N_EDGES + e];  // dst row
    atomAddF32(&deg[d], 1.0f);
  }
}

__global__ void k_rsqrt_deg(float* __restrict__ deg) {
  int i = blockIdx.x * blockDim.x + threadIdx.x;
  if (i < N_NODES) deg[i] = rsqrtf(deg[i]);  // deg becomes dinv
}

// ---------------------------------------------------------------- h = x @ W via V_WMMA_F32_16X16X4_F32
// One wave per 16-row M tile; N = 32 -> two 16x16 accumulators; K = 64 -> 16 steps of 4.
__global__ void k_gemm_xw(const float* __restrict__ x, const float* __restrict__ W,
                          float* __restrict__ h) {
  const int wavesPerBlock = blockDim.x >> 5;
  const int wave = blockIdx.x * wavesPerBlock + (threadIdx.x >> 5);
  const int lane = threadIdx.x & 31;
  const int NTILES = N_NODES / 16;           // 62500, exact
  if (wave >= NTILES) return;                // wave-uniform: EXEC stays all-1s

  const int m0   = wave * 16;
  const int mrow = m0 + (lane & 15);
  const int half = lane >> 4;                // 0: K={0,1}, 1: K={2,3} within a k-step
  const int ncol = lane & 15;

  v8f c0 = {}; v8f c1 = {};
  #pragma unroll
  for (int kk = 0; kk < IN_DIM / 4; ++kk) {
    const int kA = kk * 4 + half * 2;
    // A fragment: row mrow, cols kA, kA+1 (8B aligned vector load)
    v2f a = *(const v2f*)(x + (long long)mrow * IN_DIM + kA);
    // B fragments: rows kA,kA+1 of W, col = ncol (tile 0) / ncol+16 (tile 1)
    v2f b0, b1;
    b0[0] = W[kA * EMB + ncol];           b0[1] = W[(kA + 1) * EMB + ncol];
    b1[0] = W[kA * EMB + 16 + ncol];      b1[1] = W[(kA + 1) * EMB + 16 + ncol];
    c0 = __builtin_amdgcn_wmma_f32_16x16x4_f32(false, a, false, b0, (short)0, c0, false, false);
    c1 = __builtin_amdgcn_wmma_f32_16x16x4_f32(false, a, false, b1, (short)0, c1, false, false);
  }

  // C/D layout: lanes 0-15 -> M = r, lanes 16-31 -> M = 8 + r; N = lane&15
  const int mbase = m0 + half * 8;
  #pragma unroll
  for (int r = 0; r < 8; ++r) {
    h[(long long)(mbase + r) * EMB + ncol]      = c0[r];
    h[(long long)(mbase + r) * EMB + 16 + ncol] = c1[r];
  }
}

// ---------------------------------------------------------------- edge scatter: agg[dst] += h[src] * (dinv[s]*dinv[d])
// One thread per (edge, 4-float chunk): 128-bit gather + 4 native f32 atomics.
__global__ void k_edge_scatter(const long long* __restrict__ ei,
                               const float* __restrict__ h,
                               const float* __restrict__ dinv,
                               float* __restrict__ agg) {
  long long idx = (long long)blockIdx.x * blockDim.x + threadIdx.x;
  if (idx >= (long long)N_EDGES * (EMB / 4)) return;
  const int e  = (int)(idx >> 3);
  const int c4 = (int)(idx & 7) << 2;
  const int s = (int)ei[e];
  const int d = (int)ei[(long long)N_EDGES + e];
  const float nrm = dinv[s] * dinv[d];
  const float4 v = *(const float4*)(h + (long long)s * EMB + c4);
  float* ap = agg + (long long)d * EMB + c4;
  atomAddF32(ap + 0, v.x * nrm);
  atomAddF32(ap + 1, v.y * nrm);
  atomAddF32(ap + 2, v.z * nrm);
  atomAddF32(ap + 3, v.w * nrm);
}

// ---------------------------------------------------------------- self-loop + bias + tanh + mean-pool scatter
__global__ void k_self_pool(const float* __restrict__ h, const float* __restrict__ agg,
                            const float* __restrict__ dinv, const float* __restrict__ bgcn,
                            const long long* __restrict__ batch,
                            float* __restrict__ pooled, float* __restrict__ cnt,
                            int colOff) {
  int idx = blockIdx.x * blockDim.x + threadIdx.x;
  int node = idx >> 5;                       // 32 lanes per node (wave32-friendly)
  int j = idx & 31;
  if (node >= N_NODES) return;
  float di = dinv[node];
  long long o = (long long)node * EMB + j;
  float v = agg[o] + h[o] * (di * di) + bgcn[j];   // 1/deg == dinv^2
  v = tanhf(v);
  int g = (int)batch[node];
  atomAddF32(&pooled[(long long)g * (2 * EMB) + colOff + j], v);
  if (j == 0) atomAddF32(&cnt[g], 1.0f);
}

// ---------------------------------------------------------------- final: out = tanh(pooled/cnt) @ W_out + b_out
__global__ void k_classifier(const float* __restrict__ pooled,
                             const float* __restrict__ cnt_s, const float* __restrict__ cnt_t,
                             const float* __restrict__ Wout, const float* __restrict__ bout,
                             float* __restrict__ out) {
  __shared__ float pv[2 * EMB];
  const int g = blockIdx.x;
  const int t = threadIdx.x;
  if (t < 2 * EMB) {
    float c = (t < EMB) ? cnt_s[g] : cnt_t[g];
    c = fmaxf(c, 1.0f);
    pv[t] = tanhf(pooled[(long long)g * (2 * EMB) + t] / c);
  }
  __syncthreads();
  for (int c = t; c < NUM_CLASSES; c += blockDim.x) {
    float acc = bout[c];
    #pragma unroll
    for (int j = 0; j < 2 * EMB; ++j)
      acc = fmaf(pv[j], Wout[j * NUM_CLASSES + c], acc);
    out[(long long)g * NUM_CLASSES + c] = acc;
  }
}

// ---------------------------------------------------------------- host
extern "C" void kernel_launch(void* const* d_in, const int* in_sizes, int n_in,
                              void* d_out, int out_size, void* d_ws, size_t ws_size,
                              hipStream_t stream) {
  (void)in_sizes; (void)n_in; (void)out_size; (void)ws_size;

  const float*     x_s   = (const float*)d_in[0];
  const long long* ei_s  = (const long long*)d_in[1];
  const long long* b_s   = (const long long*)d_in[2];
  const float*     x_t   = (const float*)d_in[3];
  const long long* ei_t  = (const long long*)d_in[4];
  const long long* b_t   = (const long long*)d_in[5];
  // d_in[6] = y (unused by forward)
  const float*     W_gcn = (const float*)d_in[7];
  const float*     b_gcn = (const float*)d_in[8];
  const float*     W_out = (const float*)d_in[9];
  const float*     b_out = (const float*)d_in[10];
  float* out = (float*)d_out;

  // workspace carve-up (256B aligned)
  char* ws = (char*)d_ws;
  size_t off = 0;
  auto carve = [&](size_t bytes) { char* p = ws + off; off = (off + bytes + 255) & ~(size_t)255; return p; };
  float* deg    = (float*)carve(sizeof(float) * N_NODES);               // becomes dinv
  float* h      = (float*)carve(sizeof(float) * (size_t)N_NODES * EMB); // 128 MB
  float* agg    = (float*)carve(sizeof(float) * (size_t)N_NODES * EMB); // 128 MB
  float* pooled = (float*)carve(sizeof(float) * NUM_GRAPHS * 2 * EMB);  // 8 MB
  float* cnt_s  = (float*)carve(sizeof(float) * NUM_GRAPHS);
  float* cnt_t  = (float*)carve(sizeof(float) * NUM_GRAPHS);

  const int B = 256;
  const int gNodes   = (N_NODES + B - 1) / B;
  const int gEdges   = (N_EDGES + B - 1) / B;
  const int gNodeEmb = ((N_NODES * EMB) + B - 1) / B;             // 32M threads
  const int gScatter = (int)(((long long)N_EDGES * (EMB / 4) + B - 1) / B);
  const int gGemm    = ((N_NODES / 16) + (B / 32) - 1) / (B / 32);

  // zero pooled + counts (fresh every call: determinism)
  k_zero<<<(NUM_GRAPHS * 2 * EMB + B - 1) / B, B, 0, stream>>>(pooled, NUM_GRAPHS * 2 * EMB);
  k_zero<<<(NUM_GRAPHS + B - 1) / B, B, 0, stream>>>(cnt_s, NUM_GRAPHS);
  k_zero<<<(NUM_GRAPHS + B - 1) / B, B, 0, stream>>>(cnt_t, NUM_GRAPHS);

  for (int side = 0; side < 2; ++side) {
    const float*     x     = side == 0 ? x_s : x_t;
    const long long* ei    = side == 0 ? ei_s : ei_t;
    const long long* batch = side == 0 ? b_s : b_t;
    float*           cnt   = side == 0 ? cnt_s : cnt_t;

    k_init_deg<<<gNodes, B, 0, stream>>>(deg);
    k_deg_accum<<<gEdges, B, 0, stream>>>(ei, deg);
    k_rsqrt_deg<<<gNodes, B, 0, stream>>>(deg);                   // deg -> dinv
    k_zero<<<gNodeEmb, B, 0, stream>>>(agg, N_NODES * EMB);
    k_gemm_xw<<<gGemm, B, 0, stream>>>(x, W_gcn, h);              // WMMA f32 16x16x4
    k_edge_scatter<<<gScatter, B, 0, stream>>>(ei, h, deg, agg);
    k_self_pool<<<gNodeEmb, B, 0, stream>>>(h, agg, deg, b_gcn, batch, pooled, cnt, side * EMB);
  }

  k_classifier<<<NUM_GRAPHS, 128, 0, stream>>>(pooled, cnt_s, cnt_t, W_out, b_out, out);
}